// Voxelization_4148938408192
// MI455X (gfx1250) — compile-verified
//
#include <hip/hip_runtime.h>
#include <math.h>

#define BATCH 8
#define CHAN  64
#define NPTS  65536
#define RR    32
#define R3    32768            // 32^3
#define CHUNK 2048             // points staged per async chunk
#define NCHUNK (NPTS / CHUNK)  // 32

#ifndef __has_builtin
#define __has_builtin(x) 0
#endif

#if __has_builtin(__builtin_amdgcn_global_load_async_to_lds_b64)
#define HAVE_ASYNC_B64 1
#else
#define HAVE_ASYNC_B64 0
#endif
#if __has_builtin(__builtin_amdgcn_s_wait_asynccnt)
#define HAVE_WAIT_ASYNC 1
#else
#define HAVE_WAIT_ASYNC 0
#endif

typedef int v2i __attribute__((vector_size(8)));   // 2 x i32, 8 bytes

// Async DMA: copy 8 bytes / lane from global memory into LDS (ASYNCcnt-tracked).
// Builtin signature (from clang diagnostic): (v2i addrspace(1)*, v2i addrspace(3)*, Imm, Imm)
__device__ __forceinline__ void async_copy_b64(const void* gsrc, void* ldst) {
#if HAVE_ASYNC_B64
  __builtin_amdgcn_global_load_async_to_lds_b64(
      (__attribute__((address_space(1))) v2i*)gsrc,
      (__attribute__((address_space(3))) v2i*)ldst, 0, 0);
#else
  asm volatile("global_load_async_to_lds_b64 %0, %1, off"
               :: "v"((unsigned)(uintptr_t)ldst), "v"(gsrc)
               : "memory");
#endif
}

#if HAVE_WAIT_ASYNC
#define WAIT_ASYNC(n) __builtin_amdgcn_s_wait_asynccnt(n)
#else
#define WAIT_ASYNC(n) asm volatile("s_wait_asynccnt %0" :: "i"(n) : "memory")
#endif

// ---------------------------------------------------------------------------
// Kernel 1: per-batch coordinate statistics (deterministic tree reductions).
// stats[b*4 + {0,1,2}] = mean(x,y,z); stats[b*4+3] = denom = 2*sqrt(max||.||^2)
// ---------------------------------------------------------------------------
__global__ void __launch_bounds__(1024) stats_kernel(const float* __restrict__ coords,
                                                     float* __restrict__ stats) {
  __shared__ float red[3 * 1024];
  const int b = blockIdx.x;
  const int t = threadIdx.x;
  const float* cx = coords + (size_t)b * 3 * NPTS;
  const float* cy = cx + NPTS;
  const float* cz = cy + NPTS;

  float sx = 0.f, sy = 0.f, sz = 0.f;
  for (int n = t; n < NPTS; n += 1024) { sx += cx[n]; sy += cy[n]; sz += cz[n]; }
  red[t] = sx; red[1024 + t] = sy; red[2048 + t] = sz;
  __syncthreads();
  for (int s = 512; s > 0; s >>= 1) {
    if (t < s) {
      red[t]        += red[t + s];
      red[1024 + t] += red[1024 + t + s];
      red[2048 + t] += red[2048 + t + s];
    }
    __syncthreads();
  }
  const float inv_n = 1.0f / (float)NPTS;
  const float mx = red[0] * inv_n;
  const float my = red[1024] * inv_n;
  const float mz = red[2048] * inv_n;
  __syncthreads();

  float m = 0.f;
  for (int n = t; n < NPTS; n += 1024) {
    float dx = cx[n] - mx, dy = cy[n] - my, dz = cz[n] - mz;
    m = fmaxf(m, dx * dx + dy * dy + dz * dz);
  }
  red[t] = m;
  __syncthreads();
  for (int s = 512; s > 0; s >>= 1) {
    if (t < s) red[t] = fmaxf(red[t], red[t + s]);
    __syncthreads();
  }
  if (t == 0) {
    stats[b * 4 + 0] = mx;
    stats[b * 4 + 1] = my;
    stats[b * 4 + 2] = mz;
    stats[b * 4 + 3] = 2.0f * sqrtf(red[0]);   // EPS == 0
  }
}

// ---------------------------------------------------------------------------
// Kernel 2: zero per-voxel counts (workspace is poisoned).
// ---------------------------------------------------------------------------
__global__ void zero_counts_kernel(float* __restrict__ counts) {
  int i = blockIdx.x * blockDim.x + threadIdx.x;
  if (i < BATCH * R3) counts[i] = 0.0f;
}

// ---------------------------------------------------------------------------
// Kernel 3: normalize coords -> nc (2nd output), flat voxel id, counts.
// rintf == round-half-even, matching jnp.round. Counts are integer-valued so
// float atomic ordering cannot change them.
// ---------------------------------------------------------------------------
__global__ void __launch_bounds__(256) points_kernel(const float* __restrict__ coords,
                                                     const float* __restrict__ stats,
                                                     float* __restrict__ nc_out,
                                                     int* __restrict__ ids,
                                                     float* __restrict__ counts) {
  int t = blockIdx.x * blockDim.x + threadIdx.x;
  if (t >= BATCH * NPTS) return;
  const int b = t >> 16;            // / NPTS
  const int n = t & (NPTS - 1);
  const float* cb = coords + (size_t)b * 3 * NPTS;
  float*       nb = nc_out + (size_t)b * 3 * NPTS;
  const float inv_d = 1.0f / stats[b * 4 + 3];

  int q[3];
#pragma unroll
  for (int d = 0; d < 3; ++d) {
    float v = (cb[d * NPTS + n] - stats[b * 4 + d]) * inv_d + 0.5f;
    v = fminf(fmaxf(v * (float)RR, 0.0f), (float)(RR - 1));
    nb[d * NPTS + n] = v;
    q[d] = (int)rintf(v);
  }
  int flat = (q[0] * RR + q[1]) * RR + q[2];
  ids[(size_t)b * NPTS + n] = flat;
  unsafeAtomicAdd(&counts[b * R3 + flat], 1.0f);   // -> global_atomic_add_f32
}

// ---------------------------------------------------------------------------
// Kernel 4: scatter-mean with full LDS privatization + async double-buffered
// staging. One workgroup owns (batch, channel-pair): 256 KB accumulator plus
// 2 x 24 KB staging buffers (ids + 2 feature channels) = 304 KB of the WGP's
// 320 KB LDS. Chunks are DMA'd global->LDS (ASYNCcnt), accumulated with LDS
// atomics (ds_add_f32), then sum/max(cnt,1) is streamed out.
// ---------------------------------------------------------------------------
__global__ void __launch_bounds__(1024) scatter_kernel(const float* __restrict__ feat,
                                                       const int* __restrict__ ids,
                                                       const float* __restrict__ counts,
                                                       float* __restrict__ out) {
  __shared__ float acc[2 * R3];                  // 256 KB accumulator
  __shared__ __align__(16) int   sid[2][CHUNK];  // 16 KB staging (voxel ids)
  __shared__ __align__(16) float sf0[2][CHUNK];  // 16 KB staging (channel c0)
  __shared__ __align__(16) float sf1[2][CHUNK];  // 16 KB staging (channel c0+1)

  const int wg = blockIdx.x;        // 0 .. BATCH*(CHAN/2)-1
  const int b  = wg >> 5;           // / (CHAN/2)
  const int c0 = (wg & 31) * 2;
  const int t  = threadIdx.x;
  const int e  = t * 2;             // this lane's 2 elements within a chunk

  const int*   gid = ids + (size_t)b * NPTS;
  const float* gf0 = feat + ((size_t)b * CHAN + c0) * NPTS;
  const float* gf1 = gf0 + NPTS;

  // Prologue: launch DMA for chunk 0, overlap with accumulator zeroing.
  async_copy_b64(gid + e, &sid[0][e]);
  async_copy_b64(gf0 + e, &sf0[0][e]);
  async_copy_b64(gf1 + e, &sf1[0][e]);
  for (int i = t; i < 2 * R3; i += 1024) acc[i] = 0.0f;

  int buf = 0;
  for (int k = 0; k < NCHUNK; ++k) {
    if (k + 1 < NCHUNK) {
      const size_t g = (size_t)(k + 1) * CHUNK + e;
      async_copy_b64(gid + g, &sid[buf ^ 1][e]);
      async_copy_b64(gf0 + g, &sf0[buf ^ 1][e]);
      async_copy_b64(gf1 + g, &sf1[buf ^ 1][e]);
      WAIT_ASYNC(3);                // chunk k landed; chunk k+1 still in flight
    } else {
      WAIT_ASYNC(0);                // drain the last chunk
    }
    __syncthreads();                // all waves' DMA visible (also covers zeroing)

    int2   v  = ((const int2*)  &sid[buf][0])[t];
    float2 a0 = ((const float2*)&sf0[buf][0])[t];
    float2 a1 = ((const float2*)&sf1[buf][0])[t];
    atomicAdd(&acc[v.x],      a0.x);
    atomicAdd(&acc[v.y],      a0.y);
    atomicAdd(&acc[R3 + v.x], a1.x);
    atomicAdd(&acc[R3 + v.y], a1.y);

    __syncthreads();                // buffer fully consumed before reuse
    buf ^= 1;
  }

  const float* cnt = counts + (size_t)b * R3;
  float* o0 = out + ((size_t)b * CHAN + c0) * R3;
  float* o1 = o0 + R3;
  for (int i = t; i < R3; i += 1024) {
    float inv = 1.0f / fmaxf(cnt[i], 1.0f);
    o0[i] = acc[i] * inv;
    o1[i] = acc[R3 + i] * inv;
  }
}

// ---------------------------------------------------------------------------
// Launcher. d_out = [vox_feat: B*C*R3 floats][nc: B*3*N floats].
// d_ws = [stats: 32 floats (256 B slot)][ids: B*N ints][counts: B*R3 floats]
// ---------------------------------------------------------------------------
extern "C" void kernel_launch(void* const* d_in, const int* in_sizes, int n_in,
                              void* d_out, int out_size, void* d_ws, size_t ws_size,
                              hipStream_t stream) {
  const float* features = (const float*)d_in[0];   // [B, C, N]
  const float* coords   = (const float*)d_in[1];   // [B, 3, N]

  float* out = (float*)d_out;
  float* vox = out;                                  // [B, C, R^3]
  float* nc  = out + (size_t)BATCH * CHAN * R3;      // [B, 3, N]

  char*  ws     = (char*)d_ws;
  float* stats  = (float*)ws;                                        // 32 floats
  int*   ids    = (int*)(ws + 256);                                  // B*N ints
  float* counts = (float*)(ws + 256 + sizeof(int) * (size_t)BATCH * NPTS);

  stats_kernel<<<BATCH, 1024, 0, stream>>>(coords, stats);
  zero_counts_kernel<<<(BATCH * R3 + 255) / 256, 256, 0, stream>>>(counts);
  points_kernel<<<(BATCH * NPTS + 255) / 256, 256, 0, stream>>>(coords, stats, nc, ids, counts);
  scatter_kernel<<<BATCH * (CHAN / 2), 1024, 0, stream>>>(features, ids, counts, vox);
}